// InvariantMessagePassingTP_Old_85633057947776
// MI455X (gfx1250) — compile-verified
//
#include <hip/hip_runtime.h>
#include <stdint.h>
#include <stddef.h>

// InvariantMessagePassingTP for MI455X (gfx1250).
// Memory-bound op (~510 MB traffic, 1.2 GFLOP): one wave32 per receiver node,
// tp_weights streamed via CDNA5 async global->LDS DMA (ASYNCcnt), double-buffered.

#define N_NODES_C 10000
#define N_EDGES_C 200000
#define CH        128
#define WPB       8            // waves (nodes) per block
#define TPB       (WPB * 32)

struct __align__(16) F4 { float x, y, z, w; };

__device__ __forceinline__ void fma4(F4& a, float s, const F4& b) {
  a.x = __builtin_fmaf(s, b.x, a.x);
  a.y = __builtin_fmaf(s, b.y, a.y);
  a.z = __builtin_fmaf(s, b.z, a.z);
  a.w = __builtin_fmaf(s, b.w, a.w);
}

__device__ __forceinline__ F4 mul4(const F4& a, const F4& b) {
  F4 r; r.x = a.x * b.x; r.y = a.y * b.y; r.z = a.z * b.z; r.w = a.w * b.w; return r;
}

// Uniform (wave-scalar) lower_bound over the sorted int64 receiver list.
__device__ __forceinline__ int lower_bound_i64(const long long* __restrict__ r,
                                               long long key) {
  int base = 0, len = N_EDGES_C;
  while (len > 0) {
    int half = len >> 1;
    if (r[base + half] < key) { base += half + 1; len -= half + 1; }
    else                      { len = half; }
  }
  return base;
}

// Stage one edge's tp_weights block (4 x 128 f32 = 2048 B) into LDS via the
// CDNA5 async global->LDS DMA path. 32 lanes x 16 B x 4 instructions.
// NOTE: the instruction offset is added to BOTH the LDS and global address,
// so one immediate advances both sides in lockstep.
__device__ __forceinline__ void stage_tpw(const float* __restrict__ tp_e,
                                          uint32_t lds_off, int lane) {
  uint32_t voff = (uint32_t)(lane * 16);   // per-lane global offset
  uint32_t ldst = lds_off + voff;          // per-lane LDS destination
  asm volatile(
      "global_load_async_to_lds_b128 %0, %1, %2\n\t"
      "global_load_async_to_lds_b128 %0, %1, %2 offset:512\n\t"
      "global_load_async_to_lds_b128 %0, %1, %2 offset:1024\n\t"
      "global_load_async_to_lds_b128 %0, %1, %2 offset:1536"
      :: "v"(ldst), "v"(voff), "s"(tp_e)
      : "memory");
}

__global__ __launch_bounds__(TPB) void impt_kernel(
    const float* __restrict__ node_feats,    // [N_NODES, 128]
    const float* __restrict__ edge_attrs,    // [N_EDGES, 16]
    const float* __restrict__ tp_weights,    // [N_EDGES, 4, 128]
    const long long* __restrict__ sender,    // [N_EDGES]
    const long long* __restrict__ receiver,  // [N_EDGES] sorted
    float* __restrict__ out)                 // [N_NODES, 16, 128]
{
  __shared__ __align__(16) float stage[WPB][2][4 * CH];   // 32 KB / block

  const int lane = threadIdx.x & 31;
  const int wave = threadIdx.x >> 5;
  const int node = blockIdx.x * WPB + wave;
  if (node >= N_NODES_C) return;

  // Contiguous edge range owned by this node (receivers are sorted).
  const int lo = lower_bound_i64(receiver, (long long)node);
  const int hi = lower_bound_i64(receiver, (long long)node + 1);
  const int ne = hi - lo;

  F4 acc[16];
#pragma unroll
  for (int i = 0; i < 16; ++i) { acc[i].x = 0.f; acc[i].y = 0.f; acc[i].z = 0.f; acc[i].w = 0.f; }

  const uint32_t lds_base0 = (uint32_t)(uintptr_t)&stage[wave][0][0];
  const uint32_t lds_base1 = (uint32_t)(uintptr_t)&stage[wave][1][0];

  if (ne > 0) stage_tpw(tp_weights + (size_t)lo * (4 * CH), lds_base0, lane);

  for (int i = 0; i < ne; ++i) {
    const int e  = lo + i;
    const int cb = i & 1;
    const bool more = (i + 1) < ne;

    if (more) {
      // Close ds-read vs async-write race on the buffer we are about to refill,
      // then prefetch edge e+1 and wait for edge e's 4 DMA ops (in-order).
      asm volatile("s_wait_dscnt 0" ::: "memory");
      stage_tpw(tp_weights + (size_t)(e + 1) * (4 * CH),
                cb ? lds_base0 : lds_base1, lane);
      asm volatile("s_wait_asynccnt 4" ::: "memory");
    } else {
      asm volatile("s_wait_asynccnt 0" ::: "memory");
    }

    // Uniform scalar loads: sender index + 16 edge attrs.
    const long long s = sender[e];
    const float* ea = edge_attrs + (size_t)e * 16;
    const F4 a0 = *(const F4*)(ea + 0);
    const F4 a1 = *(const F4*)(ea + 4);
    const F4 a2 = *(const F4*)(ea + 8);
    const F4 a3 = *(const F4*)(ea + 12);

    // Gather sender features (5.1 MB table -> L2-resident).
    const F4 sf = *(const F4*)(node_feats + (size_t)s * CH + lane * 4);

    // tp_weights for this edge from LDS (staged by DMA).
    const float* wbuf = &stage[wave][cb][0];
    const F4 sw0 = mul4(sf, *(const F4*)(wbuf + 0 * CH + lane * 4));
    const F4 sw1 = mul4(sf, *(const F4*)(wbuf + 1 * CH + lane * 4));
    const F4 sw2 = mul4(sf, *(const F4*)(wbuf + 2 * CH + lane * 4));
    const F4 sw3 = mul4(sf, *(const F4*)(wbuf + 3 * CH + lane * 4));

    // L_OF_LM = [0, 1,1,1, 2,2,2,2,2, 3,3,3,3,3,3,3]
    fma4(acc[0],  a0.x, sw0);
    fma4(acc[1],  a0.y, sw1); fma4(acc[2],  a0.z, sw1); fma4(acc[3],  a0.w, sw1);
    fma4(acc[4],  a1.x, sw2); fma4(acc[5],  a1.y, sw2); fma4(acc[6],  a1.z, sw2);
    fma4(acc[7],  a1.w, sw2); fma4(acc[8],  a2.x, sw2);
    fma4(acc[9],  a2.y, sw3); fma4(acc[10], a2.z, sw3); fma4(acc[11], a2.w, sw3);
    fma4(acc[12], a3.x, sw3); fma4(acc[13], a3.y, sw3); fma4(acc[14], a3.z, sw3);
    fma4(acc[15], a3.w, sw3);
  }

  // Exclusive ownership of this node's output: plain b128 stores (covers the
  // zero-edge case too, so no separate zero-init pass is needed).
  float* op = out + (size_t)node * (16 * CH) + lane * 4;
#pragma unroll
  for (int lm = 0; lm < 16; ++lm) {
    *(F4*)(op + lm * CH) = acc[lm];
  }
}

extern "C" void kernel_launch(void* const* d_in, const int* in_sizes, int n_in,
                              void* d_out, int out_size, void* d_ws, size_t ws_size,
                              hipStream_t stream) {
  (void)in_sizes; (void)n_in; (void)out_size; (void)d_ws; (void)ws_size;
  const float*     node_feats = (const float*)d_in[0];
  const float*     edge_attrs = (const float*)d_in[1];
  const float*     tp_weights = (const float*)d_in[2];
  const long long* sender     = (const long long*)d_in[3];
  const long long* receiver   = (const long long*)d_in[4];
  float*           out        = (float*)d_out;

  dim3 grid((N_NODES_C + WPB - 1) / WPB);
  impt_kernel<<<grid, TPB, 0, stream>>>(node_feats, edge_attrs, tp_weights,
                                        sender, receiver, out);
}